// HitdlrLayer_77996606095564
// MI455X (gfx1250) — compile-verified
//
#include <hip/hip_runtime.h>
#include <hip/hip_bf16.h>

// ---------------------------------------------------------------------------
// HitdlrLayer forward kinematics for MI455X (gfx1250, wave32).
//
// Phase 1: per-(batch,finger) 4x4 matrix chains (scalar VALU, trivial cost).
// Phase 2: vertex transforms via V_WMMA_F32_16X16X4_F32:
//   A(16x4)  = 4 stacked 4x4 transform matrices (4 batch/finger slots)
//   B(4x16)  = 16 homogeneous vertices, column-major (v^T)
//   D(16x16) = 4 blocks of (4x16) transformed vertices; keep rows x,y,z.
// Store-bandwidth bound (~884 MB out -> ~38 us floor @ 23.3 TB/s), so each
// wave processes a strip of 8 vertex tiles with one A fragment held in
// registers: 8 WMMAs per wave amortize all addressing overhead, and the
// unrolled body gives b64-load / wmma / b96-store ILP.
// ---------------------------------------------------------------------------

typedef float v2f __attribute__((ext_vector_type(2)));
typedef float v8f __attribute__((ext_vector_type(8)));

#define TPW 8  // vertex tiles (16 verts each) per wave strip

// Finger mount frames (row-major 4x4 per finger).
__constant__ float c_TFR[5 * 16] = {
    0.429052f, -0.571046f, -0.699872f, 0.061f,
    0.187171f,  0.814201f, -0.549586f, 0.044f,
    0.883675f,  0.104806f,  0.456218f, 0.0885f,
    0.f, 0.f, 0.f, 1.f,

    0.f, -0.087156f,  0.996195f, -0.001429881f,
    0.f, -0.996195f, -0.087156f,  0.036800135f,
    1.f,  0.f,        0.f,        0.116743545f,
    0.f, 0.f, 0.f, 1.f,

    0.f,  0.f, 1.f, -0.0026f,
    0.f, -1.f, 0.f,  0.01f,
    1.f,  0.f, 0.f,  0.127043545f,
    0.f, 0.f, 0.f, 1.f,

    0.f,  0.087152f,  0.996195f, -0.001429881f,
    0.f, -0.996195f,  0.087152f, -0.016800135f,
    1.f,  0.f,        0.f,        0.122043545f,
    0.f, 0.f, 0.f, 1.f,

    0.f,  0.1736479f, 0.9848078f,  0.002071571f,
    0.f, -0.9848078f, 0.1736479f, -0.043396306f,
    1.f,  0.f,        0.f,         0.103043545f,
    0.f, 0.f, 0.f, 1.f,
};

// cos/sin of float32(pi/2) exactly as numpy float32 computes them.
#define CA12 (-4.37113883e-08f)
#define SA12 (1.0f)

__device__ __forceinline__ void mm4(const float* __restrict__ A,
                                    const float* __restrict__ B,
                                    float* __restrict__ C) {
#pragma unroll
  for (int i = 0; i < 4; ++i)
#pragma unroll
    for (int j = 0; j < 4; ++j) {
      float s = 0.f;
#pragma unroll
      for (int k = 0; k < 4; ++k) s = fmaf(A[i * 4 + k], B[k * 4 + j], s);
      C[i * 4 + j] = s;
    }
}

// DH transform with D=0: rows {c,-s,0,A | s*ca,c*ca,-sa,0 | s*sa,c*sa,ca,0 | 0,0,0,1}
__device__ __forceinline__ void dh4(float Aa, float ca, float sa, float th,
                                    float* __restrict__ M) {
  float c = __cosf(th), s = __sinf(th);
  M[0] = c;      M[1] = -s;     M[2] = 0.f;  M[3] = Aa;
  M[4] = s * ca; M[5] = c * ca; M[6] = -sa;  M[7] = 0.f;
  M[8] = s * sa; M[9] = c * sa; M[10] = ca;  M[11] = 0.f;
  M[12] = 0.f;   M[13] = 0.f;   M[14] = 0.f; M[15] = 1.f;
}

// Phase 1: one thread per (batch, finger). Workspace layout (floats):
//   ws[0 .. B*16)                          poseT (per batch)
//   ws[B*16 + s*B*5*16 + (b*5+f)*16 ..)    P_{s+1}, s = 0..3
__global__ void build_mats_kernel(const float* __restrict__ pose,
                                  const float* __restrict__ theta,
                                  const float* __restrict__ T,
                                  float* __restrict__ ws, int B) {
  int idx = blockIdx.x * blockDim.x + threadIdx.x;
  if (idx >= B * 5) return;
  int b = idx / 5;
  int f = idx - b * 5;

  float Pb[16], Tm[16], PT[16];
#pragma unroll
  for (int i = 0; i < 16; ++i) {
    Pb[i] = pose[(size_t)b * 16 + i];
    Tm[i] = T[i];
  }
  mm4(Pb, Tm, PT);
  if (f == 0) {
#pragma unroll
    for (int i = 0; i < 16; ++i) ws[(size_t)b * 16 + i] = PT[i];
  }

  float Tw0[16];
  mm4(PT, &c_TFR[f * 16], Tw0);

  float th0 = theta[(size_t)b * 20 + f * 4 + 0];
  float th1 = theta[(size_t)b * 20 + f * 4 + 1];
  float th2 = theta[(size_t)b * 20 + f * 4 + 2];
  float th3 = theta[(size_t)b * 20 + f * 4 + 3];

  float T01[16], T12[16], T23[16], T34[16];
  dh4(0.f,    1.f,  0.f,  -th0, T01);
  dh4(0.f,    CA12, SA12,  th1, T12);
  dh4(0.055f, 1.f,  0.f,   th2, T23);
  dh4(0.025f, 1.f,  0.f,   th3, T34);

  float P1[16], P2[16], P3[16], P4[16];
  mm4(Tw0, T01, P1);
  mm4(P1, T12, P2);
  mm4(P2, T23, P3);
  mm4(P3, T34, P4);

  size_t stage = (size_t)B * 5 * 16;
  size_t base = (size_t)B * 16 + (size_t)idx * 16;
#pragma unroll
  for (int i = 0; i < 16; ++i) {
    ws[base + 0 * stage + i] = P1[i];
    ws[base + 1 * stage + i] = P2[i];
    ws[base + 2 * stage + i] = P3[i];
    ws[base + 3 * stage + i] = P4[i];
  }
}

__device__ __forceinline__ v8f wmma4(v2f a, v2f b) {
  v8f c = {};
  return __builtin_amdgcn_wmma_f32_16x16x4_f32(
      /*neg_a=*/false, a, /*neg_b=*/false, b,
      /*c_mod=*/(short)0, c, /*reuse_a=*/false, /*reuse_b=*/false);
}

// Phase 2: one wave per (4-matrix group, strip of TPW 16-vertex tiles).
// A-matrix VGPR layout (16x4 f32, ISA 7.12.2): lanes 0-15 hold M=0..15 with
// {VGPR0,VGPR1} = K{0,1}; lanes 16-31 = K{2,3}. B (4x16) mirrors: VGPR0/1 hold
// rows K = half*2 / half*2+1, N = lane%16. D (16x16 f32): VGPR r holds M=r
// (lanes 0-15) / M=r+8 (lanes 16-31), N = lane%16.
__global__ __launch_bounds__(256)
void wmma_xform_kernel(const float* __restrict__ mats, int nMats,
                       const float* __restrict__ verts, int nVerts,
                       float* __restrict__ out, int nStrips, int nVTiles,
                       int totalWaves) {
  int gtid = blockIdx.x * blockDim.x + threadIdx.x;
  int wid = gtid >> 5;
  if (wid >= totalWaves) return;  // wave-uniform: EXEC stays all-ones for WMMA

  int lane = threadIdx.x & 31;
  int mg = wid / nStrips;
  int strip = wid - mg * nStrips;
  int half = lane >> 4;  // 0 -> K{0,1}, 1 -> K{2,3}
  int l = lane & 15;

  // A fragment: stacked row m = l -> matrix mg*4+(l>>2), row (l&3), cols k..k+1
  int miA = min(mg * 4 + (l >> 2), nMats - 1);
  v2f a = *(const v2f*)(mats + (size_t)miA * 16 + (size_t)(l & 3) * 4 + half * 2);

  int vt0 = strip * TPW;
  int n0 = vt0 * 16 + l;
  int mi0 = mg * 4 + half * 2;  // this lane's two result matrices: mi0, mi0+1

  bool fullStrip = ((vt0 + TPW) * 16 <= nVerts) && (mg * 4 + 3 < nMats);
  if (fullStrip) {  // wave-uniform branch
    const float* bp = verts + (size_t)n0 * 4 + half * 2;
    float* o0 = out + ((size_t)mi0 * nVerts + n0) * 3;
    float* o1 = o0 + (size_t)nVerts * 3;
#pragma unroll
    for (int t = 0; t < TPW; ++t) {
      v2f bv = *(const v2f*)(bp + t * 64);  // 16 verts * 4 floats per tile
      v8f d = wmma4(a, bv);
      o0[t * 48 + 0] = d[0]; o0[t * 48 + 1] = d[1]; o0[t * 48 + 2] = d[2];
      o1[t * 48 + 0] = d[4]; o1[t * 48 + 1] = d[5]; o1[t * 48 + 2] = d[6];
    }
  } else {
    for (int t = 0; t < TPW; ++t) {
      int vt = vt0 + t;
      if (vt >= nVTiles) break;  // wave-uniform
      int n = vt * 16 + l;
      int nc = min(n, nVerts - 1);
      v2f bv = *(const v2f*)(verts + (size_t)nc * 4 + half * 2);
      v8f d = wmma4(a, bv);  // EXEC still all-ones here
      if (n < nVerts) {
        if (mi0 < nMats) {
          float* o = out + ((size_t)mi0 * nVerts + n) * 3;
          o[0] = d[0]; o[1] = d[1]; o[2] = d[2];
        }
        if (mi0 + 1 < nMats) {
          float* o = out + ((size_t)(mi0 + 1) * nVerts + n) * 3;
          o[0] = d[4]; o[1] = d[5]; o[2] = d[6];
        }
      }
    }
  }
}

extern "C" void kernel_launch(void* const* d_in, const int* in_sizes, int n_in,
                              void* d_out, int out_size, void* d_ws, size_t ws_size,
                              hipStream_t stream) {
  const float* pose   = (const float*)d_in[0];
  const float* theta  = (const float*)d_in[1];
  const float* T      = (const float*)d_in[2];
  const float* rh_v   = (const float*)d_in[3];
  const float* base_v = (const float*)d_in[4];
  const float* prox_v = (const float*)d_in[5];
  const float* med_v  = (const float*)d_in[6];
  const float* dist_v = (const float*)d_in[7];

  const int B     = in_sizes[0] / 16;
  const int nRh   = in_sizes[3] / 4;
  const int nBase = in_sizes[4] / 4;
  const int nProx = in_sizes[5] / 4;
  const int nMed  = in_sizes[6] / 4;
  const int nDist = in_sizes[7] / 4;

  float* ws  = (float*)d_ws;
  float* out = (float*)d_out;

  // Phase 1: build poseT and P1..P4 in workspace.
  {
    int n1 = B * 5;
    build_mats_kernel<<<(n1 + 255) / 256, 256, 0, stream>>>(pose, theta, T, ws, B);
  }

  auto launch_xform = [&](const float* mats, int nMats, const float* verts,
                          int nVerts, float* o) {
    int groups = (nMats + 3) / 4;
    int vtiles = (nVerts + 15) / 16;
    int strips = (vtiles + TPW - 1) / TPW;
    int waves = groups * strips;
    long long threads = (long long)waves * 32;
    int block = 256;
    int grid = (int)((threads + block - 1) / block);
    wmma_xform_kernel<<<grid, block, 0, stream>>>(mats, nMats, verts, nVerts, o,
                                                  strips, vtiles, waves);
  };

  // Output tuple layout: rh, base, prox, med, dist (flat-concatenated).
  size_t offRh   = 0;
  size_t offBase = offRh + (size_t)B * nRh * 3;
  size_t offProx = offBase + (size_t)B * 5 * nBase * 3;
  size_t offMed  = offProx + (size_t)B * 5 * nProx * 3;
  size_t offDist = offMed + (size_t)B * 5 * nMed * 3;

  const float* matsPoseT = ws;
  const float* matsP = ws + (size_t)B * 16;
  size_t stage = (size_t)B * 5 * 16;

  launch_xform(matsPoseT,          B,     rh_v,   nRh,   out + offRh);
  launch_xform(matsP + 0 * stage,  B * 5, base_v, nBase, out + offBase);
  launch_xform(matsP + 1 * stage,  B * 5, prox_v, nProx, out + offProx);
  launch_xform(matsP + 2 * stage,  B * 5, med_v,  nMed,  out + offMed);
  launch_xform(matsP + 3 * stage,  B * 5, dist_v, nDist, out + offDist);
}